// ARP_G_58445914964029
// MI455X (gfx1250) — compile-verified
//
#include <hip/hip_runtime.h>
#include <math.h>

// AR(3) wrapped-Gaussian phase log-likelihood, MI455X (gfx1250, wave32).
// VALU-bound (not BW-bound): per-element cost dominated the 11-term logsumexp,
// so it is pruned to the 2 nearest wrap offsets when step=2pi/sigma is large
// enough that all other terms underflow to exactly 0 in f32 (bit-equivalent to
// the reference's f32 logsumexp). Uniform scalar fallback keeps generality.
// lane == d (d==32==warpSize) -> coalesced 128B/wave loads, rolling window of
// g and of the wrapped diffs (1 load + 2 wraps per output element).
// Cross-lane reduction via V_WMMA_F32_16X16X4_F32 (A=partials, B=ones):
// D[m][n] = acc[m]+acc[m+16], then 8-VGPR lane sum + one shfl_xor(16).

#define D_DIM   32
#define P_AR    3
#define KMAX    5
#define TWO_PI_F 6.28318530717958647692f
#define PI_F     3.14159265358979323846f
#define T_PER_WAVE 16
#define WAVES_PER_BLOCK 8

typedef __attribute__((ext_vector_type(2))) float v2f;
typedef __attribute__((ext_vector_type(8))) float v8f;

__device__ __forceinline__ float wrapf(float x) {
    // jnp.mod(x + pi, 2pi) - pi  ==  x - 2pi*floor((x+pi)/(2pi))
    return x - TWO_PI_F * floorf((x + PI_F) * (1.0f / TWO_PI_F));
}

// Sum a per-lane float across all 32 lanes of the wave using the f32 tensor
// path. A (16x4): lane L holds M=L%16, K=2*(L/16)+{0,1}; acc goes in the K-even
// slot, 0 in K-odd. B = ones(4x16) -> D[m][n] = acc[m]+acc[m+16] for all n.
// Lane n sums its 8 D VGPRs (rows 0-7 / 8-15 of column n), xor-16 add -> total.
// EXEC must be all ones at the call site (uniform control flow, full blocks).
__device__ __forceinline__ float wave_reduce_sum(float acc) {
    v2f a; a[0] = acc;  a[1] = 0.0f;
    v2f b; b[0] = 1.0f; b[1] = 1.0f;
    v8f c = {};
    v8f d = __builtin_amdgcn_wmma_f32_16x16x4_f32(
        /*neg_a=*/false, a, /*neg_b=*/false, b,
        /*c_mod=*/(short)0, c, /*reuse_a=*/false, /*reuse_b=*/false);
    float s = ((d[0] + d[1]) + (d[2] + d[3])) + ((d[4] + d[5]) + (d[6] + d[7]));
    s += __shfl_xor(s, 16, 32);
    return s; // replicated in every lane
}

__global__ void __launch_bounds__(256)
arp_lq_kernel(const float* __restrict__ g, const float* __restrict__ ar_c,
              const float* __restrict__ ar_phi, const float* __restrict__ ar_eta,
              float* __restrict__ ws, int mx, int T, int blocks_per_b)
{
    const int lane = threadIdx.x & 31;      // == d index
    const int wave = threadIdx.x >> 5;
    const int blk  = blockIdx.x;            // t tile
    const int bidx = blockIdx.y;            // batch

    // Broadcast parameters (tiny, cached).
    const float c_d   = ar_c[lane];
    const float phi0  = ar_phi[0];          // multiplies dx[t+1]
    const float phi1  = ar_phi[1];          // multiplies dx[t]
    const float eta_r = ar_eta[0];
    const float sigma = sqrtf(eta_r * eta_r);
    const float inv_sigma = 1.0f / sigma;
    const float log_norm  = __logf(sigma) + 0.5f * __logf(TWO_PI_F);
    const float step = TWO_PI_F * inv_sigma;   // spacing of z_k
    const float inv_step = sigma * (1.0f / TWO_PI_F);
    // step > 6 => 3rd-nearest term <= exp(-step^2/2) relative, underflows vs 1.0f
    const bool pruned = (step > 6.0f);

    int t0 = blk * (T_PER_WAVE * WAVES_PER_BLOCK) + wave * T_PER_WAVE;
    int t1 = t0 + T_PER_WAVE; if (t1 > T) t1 = T;

    const float* gp = g + (size_t)bidx * (size_t)mx * D_DIM + lane;

    float acc = 0.0f;
    if (t0 < t1) {
        // Rolling window: g2 = g[t+2], dx0 = wrap(g[t+1]-g[t]), dx1 = wrap(g[t+2]-g[t+1])
        float g0 = gp[(size_t)(t0 + 0) * D_DIM];
        float g1 = gp[(size_t)(t0 + 1) * D_DIM];
        float g2 = gp[(size_t)(t0 + 2) * D_DIM];
        float dx0 = wrapf(g1 - g0);
        float dx1 = wrapf(g2 - g1);
        if (pruned) {
            #pragma unroll 4
            for (int t = t0; t < t1; ++t) {
                int pft = t + 11; if (pft > mx - 4) pft = mx - 4;
                __builtin_prefetch(&gp[(size_t)pft * D_DIM], 0, 3);
                float g3 = gp[(size_t)(t + 3) * D_DIM];

                float g_pred = g2 + phi1 * dx0 + phi0 * dx1 + c_d;
                float x_err  = wrapf(g3 - g_pred);
                float a      = (x_err - c_d) * inv_sigma;   // z at k=0

                // two nearest wrap offsets (others underflow to exact 0 in f32)
                float kf = rintf(-a * inv_step);
                kf = fminf(fmaxf(kf, -(float)KMAX), (float)KMAX);
                float z0  = fmaf(kf, step, a);
                float d1  = (z0 >= 0.0f) ? -1.0f : 1.0f;
                float kf2 = kf + d1;
                kf2 = (kf2 > (float)KMAX || kf2 < -(float)KMAX) ? (kf - d1) : kf2;
                float z1  = fmaf(kf2, step, a);
                float q0  = z0 * z0, q1 = z1 * z1;
                float qmn = fminf(q0, q1), qmx = fmaxf(q0, q1);
                float e   = __expf(-0.5f * (qmx - qmn));
                acc += fmaf(-0.5f, qmn, __logf(1.0f + e));  // log_norm folded out

                float dxn = wrapf(g3 - g2);
                g2 = g3; dx0 = dx1; dx1 = dxn;
            }
        } else {
            // generic full 11-term logsumexp (small-step regime)
            #pragma unroll 2
            for (int t = t0; t < t1; ++t) {
                float g3 = gp[(size_t)(t + 3) * D_DIM];
                float g_pred = g2 + phi1 * dx0 + phi0 * dx1 + c_d;
                float x_err  = wrapf(g3 - g_pred);
                float a      = (x_err - c_d) * inv_sigma;
                float q[2 * KMAX + 1];
                float qmn = 3.4e38f;
                #pragma unroll
                for (int k = 0; k < 2 * KMAX + 1; ++k) {
                    float z = fmaf((float)(k - KMAX), step, a);
                    q[k] = z * z;
                    qmn = fminf(qmn, q[k]);
                }
                float sum = 0.0f;
                #pragma unroll
                for (int k = 0; k < 2 * KMAX + 1; ++k)
                    sum += __expf(-0.5f * (q[k] - qmn));
                acc += fmaf(-0.5f, qmn, __logf(sum));

                float dxn = wrapf(g3 - g2);
                g2 = g3; dx0 = dx1; dx1 = dxn;
            }
        }
        acc -= (float)(t1 - t0) * log_norm;   // folded normalization
    }

    // Wave total via WMMA (all 32 lanes active, uniform control flow here).
    float wtot = wave_reduce_sum(acc);

    __shared__ float smem[WAVES_PER_BLOCK];
    if (lane == 0) smem[wave] = wtot;
    __syncthreads();
    if (threadIdx.x == 0) {
        float bsum = 0.0f;
        #pragma unroll
        for (int w = 0; w < WAVES_PER_BLOCK; ++w) bsum += smem[w];
        ws[(size_t)bidx * blocks_per_b + blk] = bsum; // deterministic order
    }
}

// Deterministic final reduce: wave w sums the `parts` block-partials of batch w.
__global__ void __launch_bounds__(256)
arp_final_kernel(const float* __restrict__ ws, float* __restrict__ out,
                 int n_b, int parts)
{
    const int lane = threadIdx.x & 31;
    const int wave = threadIdx.x >> 5;   // == batch index
    float acc = 0.0f;
    if (wave < n_b) {
        for (int i = lane; i < parts; i += 32)
            acc += ws[(size_t)wave * parts + i];
    }
    float tot = wave_reduce_sum(acc);    // uniform: every wave participates
    if (wave < n_b && lane == 0) out[wave] = tot;
}

extern "C" void kernel_launch(void* const* d_in, const int* in_sizes, int n_in,
                              void* d_out, int out_size, void* d_ws, size_t ws_size,
                              hipStream_t stream)
{
    const float* g      = (const float*)d_in[0];
    const float* ar_c   = (const float*)d_in[1];
    const float* ar_phi = (const float*)d_in[2];
    const float* ar_eta = (const float*)d_in[3];
    float* ws  = (float*)d_ws;
    float* out = (float*)d_out;

    const int n_b = out_size;                         // 8
    const int mx  = in_sizes[0] / (n_b * D_DIM);      // 8192
    const int T   = mx - P_AR;                        // 8189
    const int tpb = T_PER_WAVE * WAVES_PER_BLOCK;     // 128 t per block
    const int blocks_per_b = (T + tpb - 1) / tpb;     // 64

    dim3 grid(blocks_per_b, n_b);
    arp_lq_kernel<<<grid, 256, 0, stream>>>(g, ar_c, ar_phi, ar_eta,
                                            ws, mx, T, blocks_per_b);
    arp_final_kernel<<<1, 256, 0, stream>>>(ws, out, n_b, blocks_per_b);
}